// FP_14396730376440
// MI455X (gfx1250) — compile-verified
//
#include <hip/hip_runtime.h>
#include <hip/hip_bf16.h>
#include <stdint.h>

// Problem geometry (fixed by the reference)
#define NXI   256
#define NYI   256
#define NDET  368
#define NVIEW 180
#define NS    513                  // NX + NY + 1
#define NOUT  (NDET * NVIEW)       // 66240

// Padded LDS image: 4 zero cols left, 4 right, 2 zero rows top, 2 bottom.
// Left pad of 4 keeps every interior row start 16B-aligned for B128 copies.
#define PADL 4
#define PADT 2
#define LW   (PADL + NXI + 4)      // 264 floats per row
#define LH   (PADT + NYI + 2)      // 260 rows
#define LDS_BYTES (LW * LH * 4)    // 274560 < 320KB per WGP

typedef __attribute__((ext_vector_type(2))) float f2;
typedef __attribute__((ext_vector_type(4))) int   v4i;

// ---- CDNA5 async global->LDS copy (ASYNCcnt path) -------------------------
__device__ __forceinline__ void async_copy_b128(float* lds_ptr, const float* gptr) {
#if __has_builtin(__builtin_amdgcn_global_load_async_to_lds_b128)
  __builtin_amdgcn_global_load_async_to_lds_b128(
      (__attribute__((address_space(1))) v4i*)gptr,
      (__attribute__((address_space(3))) v4i*)lds_ptr, 0, 0);
#else
  // GV form: vdst = LDS byte address, vaddr = 64-bit global address.
  uint32_t l = (uint32_t)(uintptr_t)lds_ptr;
  asm volatile("global_load_async_to_lds_b128 %0, %1, off"
               :: "v"(l), "v"(gptr) : "memory");
#endif
}

__device__ __forceinline__ void wait_async_zero() {
#if __has_builtin(__builtin_amdgcn_s_wait_asynccnt)
  __builtin_amdgcn_s_wait_asynccnt(0);
#else
  asm volatile("s_wait_asynccnt 0x0" ::: "memory");
#endif
}

// ---------------------------------------------------------------------------
extern "C" __global__ void __launch_bounds__(1024, 1)
fp_sino_kernel(const float* __restrict__ img,        // [256*256]
               const float* __restrict__ grid_pos,   // [D,V,S,2]
               const float* __restrict__ weighting,  // [D,V,S]
               float* __restrict__ out) {            // [D,V]
  extern __shared__ float lds[];                     // LW*LH padded image
  const int tid  = threadIdx.x;
  const int nthr = blockDim.x;

  // 1) Zero only the pad border (disjoint from the async-filled interior,
  //    so no ordering hazard with the async copies issued below).
  for (int i = tid; i < PADT * LW; i += nthr) {      // top 2 + bottom 2 rows
    lds[i] = 0.0f;
    lds[(PADT + NYI) * LW + i] = 0.0f;
  }
  for (int j = tid; j < NYI * 8; j += nthr) {        // 4 left + 4 right cols
    int r = PADT + (j >> 3);
    int c = j & 7;                                   // 0..7
    c = (c < 4) ? c : (c + NXI);                     // {0..3} or {260..263}
    lds[r * LW + c] = 0.0f;
  }

  // 2) Async DMA the 256KB image into the LDS interior, 16B per lane-op.
  //    16384 chunks (64 chunks of 4 floats per row).
  for (int c = tid; c < (NXI * NYI) / 4; c += nthr) {
    int y  = c >> 6;
    int x4 = (c & 63) << 2;
    async_copy_b128(&lds[(y + PADT) * LW + PADL + x4], img + y * NXI + x4);
  }
  wait_async_zero();      // drain this wave's ASYNCcnt
  __syncthreads();        // make all waves' LDS writes visible

  // 3) One wave per (det,view) output; 32 lanes stride the S dimension.
  const int lane    = tid & 31;
  const int wave    = tid >> 5;
  const int wpb     = nthr >> 5;                     // waves per block
  const int gwave   = blockIdx.x * wpb + wave;
  const int gstride = gridDim.x * wpb;

  const f2* __restrict__ gp2 = (const f2*)grid_pos;

  for (int od = gwave; od < NOUT; od += gstride) {
    const size_t base = (size_t)od * NS;
    float acc = 0.0f;
    for (int s = lane; s < NS; s += 32) {
      // Streamed exactly once -> non-temporal, keep L2 for the image.
      f2    g = __builtin_nontemporal_load(gp2 + base + s);
      float w = __builtin_nontemporal_load(weighting + base + s);
      // grid_sample(align_corners=False): ix = ((g+1)*W - 1)/2 = g*128 + 127.5
      float ix  = fmaf(g.x, 128.0f, 127.5f);
      float iy  = fmaf(g.y, 128.0f, 127.5f);
      float fx0 = floorf(ix);
      float fy0 = floorf(iy);
      float ax  = ix - fx0;
      float ay  = iy - fy0;
      int x = (int)fx0 + PADL;                       // [2, 260] by geometry
      int y = (int)fy0 + PADT;                       // [0, 258]
      x = min(max(x, 0), LW - 2);                    // safety clamp into pad
      y = min(max(y, 0), LH - 2);                    //  (pad cells hold 0)
      const float* r0 = lds + y * LW + x;
      float v00 = r0[0];
      float v01 = r0[1];
      float v10 = r0[LW];
      float v11 = r0[LW + 1];
      float vx0 = fmaf(ax, v01 - v00, v00);
      float vx1 = fmaf(ax, v11 - v10, v10);
      float val = fmaf(ay, vx1 - vx0, vx0);
      acc = fmaf(w, val, acc);
    }
    // wave32 butterfly reduction
    for (int off = 16; off > 0; off >>= 1)
      acc += __shfl_xor(acc, off, 32);
    if (lane == 0)
      out[od] = (acc == acc) ? acc : 0.0f;           // nan -> 0 like reference
  }
}

// ---------------------------------------------------------------------------
extern "C" void kernel_launch(void* const* d_in, const int* in_sizes, int n_in,
                              void* d_out, int out_size, void* d_ws, size_t ws_size,
                              hipStream_t stream) {
  (void)in_sizes; (void)n_in; (void)out_size; (void)d_ws; (void)ws_size;
  const float* img = (const float*)d_in[0];
  const float* gp  = (const float*)d_in[1];
  const float* wt  = (const float*)d_in[2];
  float*       out = (float*)d_out;

  dim3 grid(512);     // grid-stride over 66240 outputs; 1 WG per WGP (LDS-bound)
  dim3 block(1024);   // 32 waves -> 8 waves/SIMD32 for latency hiding
  hipLaunchKernelGGL(fp_sino_kernel, grid, block, LDS_BYTES, stream,
                     img, gp, wt, out);
}